// TemporalInteractionModel_64450279243825
// MI455X (gfx1250) — compile-verified
//
#include <hip/hip_runtime.h>

typedef float v2f __attribute__((ext_vector_type(2)));
typedef float v8f __attribute__((ext_vector_type(8)));

namespace {
constexpr int F    = 256;   // input features = K of GEMM1
constexpr int H1   = 256;   // temporal hidden size
constexpr int N1   = 1024;  // 4*H1 columns of W_temp
constexpr int H2   = 24;    // decoder hidden size
constexpr int N2   = 96;    // 4*H2 columns of W_dec
constexpr int ROWS = 32;    // rows of x handled per workgroup
constexpr int NTHR = 512;   // 16 waves (wave32)
}

__device__ __forceinline__ float sigm(float v) {
    return 1.0f / (1.0f + __expf(-v));
}
__device__ __forceinline__ float ftanh(float v) {
    // tanh(x) = 1 - 2/(exp(2x)+1); exact limits at +/-inf, uses v_exp_f32
    return 1.0f - 2.0f / (__expf(2.0f * v) + 1.0f);
}

__device__ __forceinline__ v8f wmma_f32(v2f a, v2f b, v8f c) {
    // D = A(16x4 f32) * B(4x16 f32) + C(16x16 f32)
    return __builtin_amdgcn_wmma_f32_16x16x4_f32(
        /*neg_a=*/false, a, /*neg_b=*/false, b,
        /*c_mod=*/(short)0, c, /*reuse_a=*/false, /*reuse_b=*/false);
}

__global__ __launch_bounds__(NTHR) void lstm2_fused_kernel(
    const float* __restrict__ x,   // [T*B, 256]
    const float* __restrict__ Wt,  // [256, 1024]
    const float* __restrict__ bt,  // [1024]
    const float* __restrict__ Wd,  // [256, 96]
    const float* __restrict__ bd,  // [96]
    float* __restrict__ out)       // [T, 12, 64, 2]
{
    __shared__ float xs[ROWS * F];    // 32 KB; reused as z2 buffer after GEMM1
    __shared__ float h1s[ROWS * H1];  // 32 KB

    const int tid   = threadIdx.x;
    const int wave  = tid >> 5;
    const int lane  = tid & 31;
    const int l15   = lane & 15;
    const int khalf = (lane >> 4) * 2;     // 0 or 2: K-pair this half-wave owns
    const long rowbase = (long)blockIdx.x * ROWS;

    // ---- stage contiguous 32x256 x-slab into LDS (32 KB) ----
    {
        const float4* xg = (const float4*)(x + rowbase * F);
        float4* xl = (float4*)xs;
        #pragma unroll
        for (int i = 0; i < (ROWS * F / 4) / NTHR; ++i)
            xl[tid + i * NTHR] = xg[tid + i * NTHR];
    }
    __syncthreads();

    // ---- GEMM1 (i/g/o blocks only; f block is dead since c0 = 0) + gates ----
    {
        const int col = (wave << 4) + l15;              // h1 column 0..255
        const float* pBi = Wt + (long)khalf * N1 + col;          // i: cols [0,256)
        const float* pBg = pBi + 2 * H1;                         // g: cols [512,768)
        const float* pBo = pBi + 3 * H1;                         // o: cols [768,1024)
        const float* pA0 = xs + l15 * F + khalf;                 // rows 0..15
        const float* pA1 = xs + (16 + l15) * F + khalf;          // rows 16..31

        v8f ai0 = {}, ag0 = {}, ao0 = {};
        v8f ai1 = {}, ag1 = {}, ao1 = {};
        #pragma unroll 4
        for (int k = 0; k < F; k += 4) {
            v2f bi, bg, bo;
            bi[0] = pBi[0]; bi[1] = pBi[N1];
            bg[0] = pBg[0]; bg[1] = pBg[N1];
            bo[0] = pBo[0]; bo[1] = pBo[N1];
            pBi += 4 * N1; pBg += 4 * N1; pBo += 4 * N1;
            v2f a0 = *(const v2f*)(pA0 + k);
            v2f a1 = *(const v2f*)(pA1 + k);
            ai0 = wmma_f32(a0, bi, ai0);
            ag0 = wmma_f32(a0, bg, ag0);
            ao0 = wmma_f32(a0, bo, ao0);
            ai1 = wmma_f32(a1, bi, ai1);
            ag1 = wmma_f32(a1, bg, ag1);
            ao1 = wmma_f32(a1, bo, ao1);
        }
        const float bis = bt[col];
        const float bgs = bt[2 * H1 + col];
        const float bos = bt[3 * H1 + col];
        const int rloc = (lane >> 4) * 8;   // C/D layout: VGPR v -> row v or v+8
        #pragma unroll
        for (int v = 0; v < 8; ++v) {
            float i0 = sigm(ai0[v] + bis);
            float g0 = ftanh(ag0[v] + bgs);
            float o0 = sigm(ao0[v] + bos);
            h1s[(rloc + v) * H1 + col] = o0 * ftanh(i0 * g0);
            float i1 = sigm(ai1[v] + bis);
            float g1 = ftanh(ag1[v] + bgs);
            float o1 = sigm(ao1[v] + bos);
            h1s[(16 + rloc + v) * H1 + col] = o1 * ftanh(i1 * g1);
        }
    }
    __syncthreads();           // h1 ready; xs is now free for reuse

    float* z2s = xs;           // 32 x 96 floats = 12 KB, aliases x-slab buffer

    // ---- GEMM2: z2 = h1 @ W_dec, 12 tiles of 16x16 across waves 0..11 ----
    if (wave < 12) {           // wave-uniform branch: EXEC stays all-ones for WMMA
        const int rt = wave / 6, ct = wave % 6;
        const int col = ct * 16 + l15;                           // z2 column 0..95
        const float* pB = Wd + (long)khalf * N2 + col;
        const float* pA = h1s + (rt * 16 + l15) * H1 + khalf;
        v8f acc = {};
        #pragma unroll 4
        for (int k = 0; k < H1; k += 4) {
            v2f b; b[0] = pB[0]; b[1] = pB[N2];
            pB += 4 * N2;
            v2f a = *(const v2f*)(pA + k);
            acc = wmma_f32(a, b, acc);
        }
        const int rloc = rt * 16 + (lane >> 4) * 8;
        #pragma unroll
        for (int v = 0; v < 8; ++v)
            z2s[(rloc + v) * N2 + col] = acc[v];
    }
    __syncthreads();

    // ---- LSTM2 gates + transposed store: out[t, n/2, b, n%2] ----
    for (int idx = tid; idx < ROWS * H2; idx += NTHR) {
        const int r = idx / H2;
        const int n = idx % H2;
        const float zi = z2s[r * N2 + n]            + bd[n];
        const float zg = z2s[r * N2 + 2 * H2 + n]   + bd[2 * H2 + n];
        const float zo = z2s[r * N2 + 3 * H2 + n]   + bd[3 * H2 + n];
        const float c  = sigm(zi) * ftanh(zg);
        const float h  = sigm(zo) * ftanh(c);
        const long grow = rowbase + r;       // global row in [0, T*B)
        const long t = grow >> 6;            // / BATCH(64)
        const long b = grow & 63;
        out[((t * 12 + (n >> 1)) * 64 + b) * 2 + (n & 1)] = h;
    }
}

extern "C" void kernel_launch(void* const* d_in, const int* in_sizes, int n_in,
                              void* d_out, int out_size, void* d_ws, size_t ws_size,
                              hipStream_t stream) {
    (void)n_in; (void)out_size; (void)d_ws; (void)ws_size;
    const float* x  = (const float*)d_in[0];   // gcn_output [T,64,256]
    const float* Wt = (const float*)d_in[1];   // W_temp [256,1024]
    const float* bt = (const float*)d_in[2];   // b_temp [1024]
    const float* Wd = (const float*)d_in[3];   // W_dec [256,96]
    const float* bd = (const float*)d_in[4];   // b_dec [96]
    float* out = (float*)d_out;

    const long rows   = (long)in_sizes[0] / F;   // T*B = 65536
    const int  blocks = (int)(rows / ROWS);      // 2048
    lstm2_fused_kernel<<<blocks, NTHR, 0, stream>>>(x, Wt, bt, Wd, bd, out);
}